// PredictionDecoder_39436389711933
// MI455X (gfx1250) — compile-verified
//
#include <hip/hip_runtime.h>
#include <hip/hip_bf16.h>
#include <stdint.h>

#define NUM_ANCH    120087
#define NUM_CLS     14
#define BATCH       16
#define ROW         18        // 4 box + 14 class floats per anchor
#define K_CAND      512
#define NCAND       1024      // candidate buffer (power of two for bitonic)
#define NBINS       8192
#define SCORE_THR   0.03f
#define IOU_THR     0.5f
#define MAX_PC      100       // per class
#define MAX_TOT     100

// ---------- ordered-float key helpers ----------
__device__ __forceinline__ uint32_t float_okey(float f) {
  uint32_t u = __float_as_uint(f);
  return (u & 0x80000000u) ? ~u : (u | 0x80000000u);
}
__device__ __forceinline__ float okey_float(uint32_t k) {
  uint32_t u = (k & 0x80000000u) ? (k ^ 0x80000000u) : ~k;
  return __uint_as_float(u);
}

// ---------- CDNA5 async global->LDS (gfx1250) ----------
// VDST vgpr carries the LDS byte address; 64-bit vaddr pair; tracked by ASYNCcnt.
__device__ __forceinline__ void async_g2l_b32(uint32_t lds_off, const void* gaddr) {
  asm volatile("global_load_async_to_lds_b32 %0, %1, off"
               :: "v"(lds_off), "v"((uint64_t)(uintptr_t)gaddr)
               : "memory");
}
__device__ __forceinline__ void wait_async_le1() {
  asm volatile("s_wait_asynccnt 0x1" ::: "memory");
}
__device__ __forceinline__ void wait_async_le0() {
  asm volatile("s_wait_asynccnt 0x0" ::: "memory");
}
__device__ __forceinline__ uint32_t lds_off32(const void* p) {
  // generic LDS address = {shared aperture, offset[31:0]}; low 32 bits are the LDS offset
  return (uint32_t)(uintptr_t)p;
}

// ================= Kernel 1: decode boxes =================
__global__ __launch_bounds__(256) void k_decode(const float* __restrict__ pred,
                                                float* __restrict__ boxes) {
  int g = blockIdx.x * 256 + threadIdx.x;
  if (g >= BATCH * NUM_ANCH) return;
  int a = g % NUM_ANCH;

  int s, base;
  if      (a <  90000) { s = 100; base = 0;      }
  else if (a < 112500) { s = 50;  base = 90000;  }
  else if (a < 118125) { s = 25;  base = 112500; }
  else if (a < 119646) { s = 13;  base = 118125; }
  else                 { s = 7;   base = 119646; }

  int rem  = a - base;
  int cell = rem / 9;
  int k    = rem - cell * 9;
  float fw = 1.0f / (float)s;
  int iy = cell / s;
  int ix = cell - iy * s;
  float cy = ((float)iy + 0.5f) * fw;
  float cx = ((float)ix + 0.5f) * fw;

  const float SC[3] = {1.0f, 1.2599210498948732f, 1.5874010519681994f};   // 2^(j/3)
  const float SR[3] = {0.7071067811865476f, 1.0f, 1.4142135623730951f};   // sqrt(ratio)
  int ks = k / 3, kr = k - ks * 3;
  float aw = SC[ks] * SR[kr] * fw;
  float ah = SC[ks] / SR[kr] * fw;

  const float* p = pred + (size_t)g * ROW;
  float bx = p[0] * 0.1f, by = p[1] * 0.1f;
  float bw = p[2] * 0.2f, bh = p[3] * 0.2f;
  float ox = bx * aw + cx;
  float oy = by * ah + cy;
  float ow = __expf(bw) * aw;
  float oh = __expf(bh) * ah;

  float4 out = make_float4(ox - ow * 0.5f, oy - oh * 0.5f,
                           ox + ow * 0.5f, oy + oh * 0.5f);
  *((float4*)(boxes + (size_t)g * 4)) = out;
}

// ================= Kernel 2: per (batch,class) top-512 + NMS + top-100 =================
__global__ __launch_bounds__(256) void k_nms_class(const float* __restrict__ pred,
                                                   const float* __restrict__ boxes,
                                                   float* __restrict__ pc_score,
                                                   float* __restrict__ pc_box) {
  const int b   = blockIdx.x / NUM_CLS;
  const int c   = blockIdx.x % NUM_CLS;
  const int tid = threadIdx.x;

  __shared__ uint32_t           hist[NBINS];          // 32 KB
  __shared__ float              tile[2][256];         //  2 KB (double-buffer for async)
  __shared__ unsigned long long key[NCAND];           //  8 KB
  __shared__ float              bx4[K_CAND][4];       //  8 KB
  __shared__ float              scv[K_CAND];          //  2 KB
  __shared__ float              area[K_CAND];         //  2 KB
  __shared__ int                keep[K_CAND];         //  2 KB
  __shared__ uint32_t           chunk[256];           //  1 KB
  __shared__ uint32_t           s_thresh, s_cnt;

  for (int i = tid; i < NBINS; i += 256) hist[i] = 0;
  if (tid == 0) s_cnt = 0;
  __syncthreads();

  const float* sbase = pred + (size_t)b * NUM_ANCH * ROW + 4 + c;
  const int ntiles = (NUM_ANCH + 255) / 256;

  // ---- pass A: histogram of ordered-float keys (async double-buffered staging) ----
  {
    int i0 = tid; if (i0 >= NUM_ANCH) i0 = NUM_ANCH - 1;      // clamp: keep EXEC full
    async_g2l_b32(lds_off32(&tile[0][tid]), sbase + (size_t)i0 * ROW);
    for (int t = 0; t < ntiles; ++t) {
      int nb = t + 1;
      if (nb < ntiles) {
        int idx = nb * 256 + tid; if (idx >= NUM_ANCH) idx = NUM_ANCH - 1;
        async_g2l_b32(lds_off32(&tile[nb & 1][tid]), sbase + (size_t)idx * ROW);
        wait_async_le1();                                      // tile t has landed
      } else {
        wait_async_le0();
      }
      int idx = t * 256 + tid;
      if (idx < NUM_ANCH) {
        float sv = tile[t & 1][tid];
        atomicAdd(&hist[float_okey(sv) >> 19], 1u);            // top 13 bits -> 8192 bins
      }
    }
  }
  __syncthreads();

  // ---- find threshold bin T: count(bin >= T) >= K_CAND ----
  {
    uint32_t acc = 0;
    int cb = tid * 32;
    for (int i = 0; i < 32; ++i) acc += hist[cb + i];
    chunk[tid] = acc;
  }
  __syncthreads();
  if (tid == 0) {
    uint32_t acc = 0;
    int T = 0, ch = 255;
    for (; ch >= 0; --ch) {
      if (acc + chunk[ch] >= (uint32_t)K_CAND) break;
      acc += chunk[ch];
    }
    if (ch >= 0) {
      int bin = ch * 32 + 31;
      for (; bin >= ch * 32; --bin) {
        acc += hist[bin];
        if (acc >= (uint32_t)K_CAND) break;
      }
      T = (bin < ch * 32) ? ch * 32 : bin;
    }
    s_thresh = (uint32_t)T;
  }
  __syncthreads();
  const uint32_t T = s_thresh;

  // ---- pass B: compact candidates >= threshold (async staging again) ----
  {
    int i0 = tid; if (i0 >= NUM_ANCH) i0 = NUM_ANCH - 1;
    async_g2l_b32(lds_off32(&tile[0][tid]), sbase + (size_t)i0 * ROW);
    for (int t = 0; t < ntiles; ++t) {
      int nb = t + 1;
      if (nb < ntiles) {
        int idx = nb * 256 + tid; if (idx >= NUM_ANCH) idx = NUM_ANCH - 1;
        async_g2l_b32(lds_off32(&tile[nb & 1][tid]), sbase + (size_t)idx * ROW);
        wait_async_le1();
      } else {
        wait_async_le0();
      }
      int idx = t * 256 + tid;
      if (idx < NUM_ANCH) {
        uint32_t k32 = float_okey(tile[t & 1][tid]);
        if ((k32 >> 19) >= T) {
          uint32_t pos = atomicAdd(&s_cnt, 1u);
          if (pos < NCAND)
            key[pos] = ((unsigned long long)k32 << 32) |
                       (unsigned long long)(0xFFFFFFFFu - (uint32_t)idx);
        }
      }
    }
  }
  __syncthreads();
  uint32_t cnum = s_cnt; if (cnum > NCAND) cnum = NCAND;
  for (int i = tid; i < NCAND; i += 256)
    if ((uint32_t)i >= cnum) key[i] = 0ull;
  __syncthreads();

  // ---- bitonic sort, descending (score desc, anchor index asc on ties) ----
  for (int k = 2; k <= NCAND; k <<= 1) {
    for (int j = k >> 1; j > 0; j >>= 1) {
      for (int i = tid; i < NCAND; i += 256) {
        int l = i ^ j;
        if (l > i) {
          unsigned long long a = key[i], bb = key[l];
          bool down = ((i & k) == 0);
          if (down ? (a < bb) : (a > bb)) { key[i] = bb; key[l] = a; }
        }
      }
      __syncthreads();
    }
  }

  // ---- load top K_CAND candidates' boxes / scores ----
  for (int i = tid; i < K_CAND; i += 256) {
    unsigned long long kv = key[i];
    uint32_t k32  = (uint32_t)(kv >> 32);
    uint32_t aidx = 0xFFFFFFFFu - (uint32_t)(kv & 0xFFFFFFFFu);
    float s = (kv == 0ull) ? -INFINITY : okey_float(k32);
    float x1 = 0.f, y1 = 0.f, x2 = 0.f, y2 = 0.f;
    if (kv != 0ull && aidx < (uint32_t)NUM_ANCH) {
      const float* bp = boxes + ((size_t)b * NUM_ANCH + aidx) * 4;
      x1 = bp[0]; y1 = bp[1]; x2 = bp[2]; y2 = bp[3];
    }
    scv[i] = s;
    bx4[i][0] = x1; bx4[i][1] = y1; bx4[i][2] = x2; bx4[i][3] = y2;
    area[i] = (x2 - x1) * (y2 - y1);
    keep[i] = (s > SCORE_THR) ? 1 : 0;
  }
  __syncthreads();

  // ---- greedy NMS scan (matches reference lax.scan semantics) ----
  for (int i = 0; i < K_CAND - 1; ++i) {
    __syncthreads();
    if (!keep[i]) continue;                 // uniform: all threads read same LDS value
    float ax1 = bx4[i][0], ay1 = bx4[i][1], ax2 = bx4[i][2], ay2 = bx4[i][3];
    float aa = area[i];
    for (int j = i + 1 + tid; j < K_CAND; j += 256) {
      if (!keep[j]) continue;
      float ix1 = fmaxf(ax1, bx4[j][0]);
      float iy1 = fmaxf(ay1, bx4[j][1]);
      float ix2 = fminf(ax2, bx4[j][2]);
      float iy2 = fminf(ay2, bx4[j][3]);
      float iw = fmaxf(ix2 - ix1, 0.f);
      float ih = fmaxf(iy2 - iy1, 0.f);
      float inter = iw * ih;
      float uni = aa + area[j] - inter;
      float iou = (uni > 0.f) ? inter / uni : 0.f;
      if (iou > IOU_THR) keep[j] = 0;
    }
  }
  __syncthreads();

  // ---- emit first 100 kept (sorted order == top_k of kept scores) ----
  if (tid == 0) {
    float* ps = pc_score + ((size_t)b * NUM_CLS + c) * MAX_PC;
    float* pb = pc_box   + ((size_t)b * NUM_CLS + c) * MAX_PC * 4;
    int n = 0;
    for (int i = 0; i < K_CAND && n < MAX_PC; ++i) {
      if (keep[i]) {
        ps[n] = scv[i];
        pb[n * 4 + 0] = bx4[i][0]; pb[n * 4 + 1] = bx4[i][1];
        pb[n * 4 + 2] = bx4[i][2]; pb[n * 4 + 3] = bx4[i][3];
        ++n;
      }
    }
    for (; n < MAX_PC; ++n) {
      ps[n] = -INFINITY;
      pb[n * 4 + 0] = 0.f; pb[n * 4 + 1] = 0.f; pb[n * 4 + 2] = 0.f; pb[n * 4 + 3] = 0.f;
    }
  }
}

// ================= Kernel 3: merge classes, global top-100 =================
__global__ __launch_bounds__(256) void k_combine(const float* __restrict__ pc_score,
                                                 const float* __restrict__ pc_box,
                                                 float* __restrict__ out) {
  const int b = blockIdx.x;
  const int tid = threadIdx.x;
  const int NE = NUM_CLS * MAX_PC;                  // 1400
  __shared__ unsigned long long kkey[NUM_CLS * MAX_PC];
  __shared__ float              ksc[NUM_CLS * MAX_PC];
  __shared__ uint32_t           cnt;
  if (tid == 0) cnt = 0;

  const float* ps = pc_score + (size_t)b * NE;
  for (int e = tid; e < NE; e += 256) {
    float s = ps[e];
    ksc[e] = s;
    kkey[e] = ((unsigned long long)float_okey(s) << 32) |
              (unsigned long long)(0xFFFFFFFFu - (uint32_t)e);
  }
  __syncthreads();

  float* ob = out;                                          // (B,100,4)
  float* os = out + (size_t)BATCH * MAX_TOT * 4;            // (B,100)
  float* oc = os + (size_t)BATCH * MAX_TOT;                 // (B,100)
  float* on = oc + (size_t)BATCH * MAX_TOT;                 // (B,)

  for (int e = tid; e < NE; e += 256) {
    unsigned long long mk = kkey[e];
    int rank = 0;
    for (int f = 0; f < NE; ++f) rank += (kkey[f] > mk) ? 1 : 0;
    if (rank < MAX_TOT) {
      float s = ksc[e];
      bool ok = (s > -INFINITY);
      float x1 = 0.f, y1 = 0.f, x2 = 0.f, y2 = 0.f, sv = 0.f, cv = 0.f;
      if (ok) {
        const float* pb = pc_box + ((size_t)b * NE + e) * 4;
        x1 = fminf(fmaxf(pb[0], 0.f), 1.f);
        y1 = fminf(fmaxf(pb[1], 0.f), 1.f);
        x2 = fminf(fmaxf(pb[2], 0.f), 1.f);
        y2 = fminf(fmaxf(pb[3], 0.f), 1.f);
        sv = s;
        cv = (float)(e / MAX_PC);
        atomicAdd(&cnt, 1u);
      }
      float* obp = ob + ((size_t)b * MAX_TOT + rank) * 4;
      obp[0] = x1; obp[1] = y1; obp[2] = x2; obp[3] = y2;
      os[(size_t)b * MAX_TOT + rank] = sv;
      oc[(size_t)b * MAX_TOT + rank] = cv;
    }
  }
  __syncthreads();
  if (tid == 0) on[b] = (float)cnt;
}

// ================= launcher =================
extern "C" void kernel_launch(void* const* d_in, const int* in_sizes, int n_in,
                              void* d_out, int out_size, void* d_ws, size_t ws_size,
                              hipStream_t stream) {
  (void)in_sizes; (void)n_in; (void)out_size; (void)ws_size;
  const float* pred = (const float*)d_in[0];
  float* out = (float*)d_out;

  float* boxes    = (float*)d_ws;                                      // B*A*4
  float* pc_score = boxes + (size_t)BATCH * NUM_ANCH * 4;              // B*14*100
  float* pc_box   = pc_score + (size_t)BATCH * NUM_CLS * MAX_PC;       // B*14*100*4

  const int total = BATCH * NUM_ANCH;
  k_decode<<<(total + 255) / 256, 256, 0, stream>>>(pred, boxes);
  k_nms_class<<<BATCH * NUM_CLS, 256, 0, stream>>>(pred, boxes, pc_score, pc_box);
  k_combine<<<BATCH, 256, 0, stream>>>(pc_score, pc_box, out);
}